// ReLoopBlockGroup_30468497997895
// MI455X (gfx1250) — compile-verified
//
#include <hip/hip_runtime.h>
#include <stdint.h>

typedef float v2f __attribute__((ext_vector_type(2)));
typedef float v8f __attribute__((ext_vector_type(8)));
typedef unsigned int u32x4 __attribute__((ext_vector_type(4)));
typedef int i32x4 __attribute__((ext_vector_type(4)));
typedef int i32x8 __attribute__((ext_vector_type(8)));

#define D_DIM   2048
#define N_BLK   8
#define EPS_F   1e-6f
#define INV_D   (1.0f / 2048.0f)
#define INV_SCALE (1.0f / 45.25483399593904156165403917471f)  // 1/sqrt(2048)

#if defined(__AMDGCN__) && __has_builtin(__builtin_amdgcn_tensor_load_to_lds)
#define USE_TDM 1
#else
#define USE_TDM 0
#endif

#if USE_TDM
// Issue one TDM 2D-tile load: `rows` rows of D_DIM f32, global row pitch
// `row_stride` elements, packed contiguously into LDS at byte offset lds_off.
// Descriptor layout per CDNA5 ISA ch.8 (Tensor DMA Descriptor).
__device__ __forceinline__ void tdm_load_rows(uint64_t gaddr, uint32_t lds_off,
                                              uint32_t rows, uint64_t row_stride) {
    u32x4 g0;
    g0.x = 1u;                                                    // count=1, user D#
    g0.y = lds_off;                                               // lds_addr (bytes)
    g0.z = (uint32_t)gaddr;                                       // global_addr[31:0]
    g0.w = (uint32_t)((gaddr >> 32) & 0x01FFFFFFull) | 0x80000000u; // [56:32] | type=2
    i32x8 g1;
    g1[0] = 0x20000;                                              // data_size=2 (4B)
    g1[1] = (int)(D_DIM << 16);                                   // tensor_dim0 lo16
    g1[2] = (int)(rows << 16);                                    // tensor_dim1 lo16
    g1[3] = (int)(D_DIM << 16);                                   // tile_dim0
    g1[4] = (int)rows;                                            // tile_dim1, tile_dim2=0
    g1[5] = (int)(uint32_t)row_stride;                            // dim0_stride[31:0]
    g1[6] = (int)(uint32_t)((row_stride >> 32) & 0xFFFFull);      // dim0_stride[47:32]
    g1[7] = 0;                                                    // dim1_stride (unused, 2D)
    i32x4 gz = {0, 0, 0, 0};
#if __has_include(<hip/amd_detail/amd_gfx1250_TDM.h>)
    i32x8 gz8 = {0, 0, 0, 0, 0, 0, 0, 0};
    __builtin_amdgcn_tensor_load_to_lds(g0, g1, gz, gz, gz8, 0);  // 6-arg toolchain
#else
    __builtin_amdgcn_tensor_load_to_lds(g0, g1, gz, gz, 0);       // 5-arg ROCm 7.2
#endif
}
#endif

// One workgroup per token (b,t). 256 threads = 8 waves.
// LDS rows: 0..7 completed blocks, 8 partial, 9 effective query, 10 zeros (WMMA pad).
__global__ __launch_bounds__(256)
void router_token_kernel(const float* __restrict__ completed,
                         const float* __restrict__ partial,
                         const float* __restrict__ wq,
                         const float* __restrict__ knw,
                         float* __restrict__ out_hidden,
                         float* __restrict__ out_ent,
                         float* __restrict__ out_wts,
                         int BT)
{
    __shared__ float xs[11 * D_DIM];       // 88 KB
    __shared__ float gram[4 * 256];        // 4 partial 16x16 Grams
    __shared__ float coef[12];

    const int tid = threadIdx.x;
    const int wg  = blockIdx.x;                       // token index in [0, BT)
    const size_t tokOff = (size_t)wg * D_DIM;
    float4* xs4 = (float4*)xs;

    // ---------------- stage token working set into LDS ----------------------
#if USE_TDM
    if (tid < 32) {
        const uint32_t lds_base = (uint32_t)(uintptr_t)(void*)xs;
        tdm_load_rows((uint64_t)(uintptr_t)completed + tokOff * sizeof(float),
                      lds_base, N_BLK, (uint64_t)BT * D_DIM);
        tdm_load_rows((uint64_t)(uintptr_t)partial + tokOff * sizeof(float),
                      lds_base + N_BLK * D_DIM * sizeof(float), 1, D_DIM);
    }
#else
    const float4* prow[9];
#pragma unroll
    for (int n = 0; n < N_BLK; ++n)
        prow[n] = (const float4*)(completed + (size_t)n * BT * D_DIM + tokOff);
    prow[8] = (const float4*)(partial + tokOff);
#pragma unroll
    for (int j = 0; j < 2; ++j) {
        const int i4 = tid + j * 256;
#pragma unroll
        for (int r = 0; r < 9; ++r)
            xs4[r * 512 + i4] = prow[r][i4];
    }
#endif
    // effective query row (q = w_query * key_norm_weight) and zero pad row
#pragma unroll
    for (int j = 0; j < 2; ++j) {
        const int i4 = tid + j * 256;
        float4 a = ((const float4*)wq)[i4];
        float4 g = ((const float4*)knw)[i4];
        float4 q; q.x = a.x*g.x; q.y = a.y*g.y; q.z = a.z*g.z; q.w = a.w*g.w;
        xs4[9 * 512 + i4] = q;
        float4 z; z.x = 0.f; z.y = 0.f; z.z = 0.f; z.w = 0.f;
        xs4[10 * 512 + i4] = z;
    }
#if USE_TDM
    if (tid < 32)
        __builtin_amdgcn_s_wait_tensorcnt(0);
#endif
    __syncthreads();

    // ---------------- Gram matrix via WMMA: 4 waves, K split 4x512 ----------
    // A (16x4 f32): lane l -> row (l&15), K = k0 + 2*(l>=16) + {0,1}.
    // B (4x16 f32) mirrors the striping with B[k][n] = A[n][k], so each lane's
    // A and B fragments read identical LDS values -> pass `a` for both.
    if (tid < 128) {
        const int wv   = tid >> 5;                    // 0..3
        const int lane = tid & 31;
        const int half = lane >> 4;
        const int m    = lane & 15;
        const float* row = xs + (size_t)(m < 10 ? m : 10) * D_DIM; // pad rows -> zeros
        const int kbeg = wv * 512;
        v8f acc = {};
#pragma unroll 4
        for (int k0 = kbeg; k0 < kbeg + 512; k0 += 4) {
            const int ka = k0 + (half << 1);
            v2f a; a.x = row[ka]; a.y = row[ka + 1];
            acc = __builtin_amdgcn_wmma_f32_16x16x4_f32(
                false, a, false, a, (short)0, acc, false, false);
        }
        // D layout: VGPR v, lanes 0-15 -> (M=v, N=lane); lanes 16-31 -> (M=v+8)
#pragma unroll
        for (int v = 0; v < 8; ++v)
            gram[wv * 256 + (v + 8 * half) * 16 + m] = acc[v];
    }
    __syncthreads();

    // ---------------- scalar softmax / online merge (thread 0) -------------
    if (tid == 0) {
        float sc[9];
#pragma unroll
        for (int n = 0; n < 9; ++n) {
            const float ss = gram[n * 17]       + gram[256 + n * 17]
                           + gram[512 + n * 17] + gram[768 + n * 17];      // <x_n,x_n>
            const float dt = gram[n * 16 + 9]       + gram[256 + n * 16 + 9]
                           + gram[512 + n * 16 + 9] + gram[768 + n * 16 + 9]; // <x_n,q>
            sc[n] = dt * rsqrtf(ss * INV_D + EPS_F) * INV_SCALE;
        }
        float m1 = sc[0];
#pragma unroll
        for (int n = 1; n < N_BLK; ++n) m1 = fmaxf(m1, sc[n]);
        float es[8], lse = 0.0f;
#pragma unroll
        for (int n = 0; n < N_BLK; ++n) { es[n] = __expf(sc[n] - m1); lse += es[n]; }
        const float logl = __logf(lse);
        float ent1 = 0.0f;
#pragma unroll
        for (int n = 0; n < N_BLK; ++n) {
            const float p  = es[n] / lse;
            const float lp = (sc[n] - m1) - logl;
            ent1 -= p * lp;
        }
        // online-softmax merge with partial block
        const float ps = sc[8];
        const float mm = fmaxf(m1, ps);
        const float c1 = __expf(m1 - mm);
        const float c2 = __expf(ps - mm);
        const float denom = c1 * lse + c2;
        const float w1 = c1 * lse / denom;            // phase1_weight
        const float w2 = c2 / denom;                  // partial_weight
#pragma unroll
        for (int n = 0; n < N_BLK; ++n) coef[n] = es[n] * w1;
        coef[8] = w2;
        const float a1 = fmaxf(w1, 1e-8f);
        const float a2 = fmaxf(w2, 1e-8f);
        out_ent[wg] = a1 * ent1 - a1 * __logf(a1) - a2 * __logf(a2);
        const float invl = 1.0f / lse;
#pragma unroll
        for (int n = 0; n < N_BLK; ++n) out_wts[(size_t)wg * 9 + n] = es[n] * invl * w1;
        out_wts[(size_t)wg * 9 + 8] = w2;
    }
    __syncthreads();

    // ---------------- 9-way weighted combine from LDS, b128 stores ---------
    float c[9];
#pragma unroll
    for (int r = 0; r < 9; ++r) c[r] = coef[r];
    float4* hout = (float4*)(out_hidden + tokOff);
#pragma unroll
    for (int j = 0; j < 2; ++j) {
        const int i4 = tid + j * 256;
        float4 h; h.x = 0.f; h.y = 0.f; h.z = 0.f; h.w = 0.f;
#pragma unroll
        for (int r = 0; r < 9; ++r) {
            const float4 v = xs4[r * 512 + i4];
            h.x = fmaf(c[r], v.x, h.x);
            h.y = fmaf(c[r], v.y, h.y);
            h.z = fmaf(c[r], v.z, h.z);
            h.w = fmaf(c[r], v.w, h.w);
        }
        hout[i4] = h;
    }
}

extern "C" void kernel_launch(void* const* d_in, const int* in_sizes, int n_in,
                              void* d_out, int out_size, void* d_ws, size_t ws_size,
                              hipStream_t stream) {
    const float* completed = (const float*)d_in[0];   // [8, 2, 2048, 2048] f32
    const float* partial   = (const float*)d_in[1];   // [2, 2048, 2048] f32
    const float* wq        = (const float*)d_in[2];   // [2048] f32
    const float* knw       = (const float*)d_in[3];   // [2048] f32

    const int BT = in_sizes[1] / D_DIM;               // B*T = 4096 tokens

    float* out        = (float*)d_out;
    float* out_hidden = out;                          // BT*D floats
    float* out_ent    = out + (size_t)BT * D_DIM;     // BT floats
    float* out_wts    = out_ent + BT;                 // BT*9 floats

    router_token_kernel<<<BT, 256, 0, stream>>>(
        completed, partial, wq, knw, out_hidden, out_ent, out_wts, BT);
}